// SAGEConv_28991029248362
// MI455X (gfx1250) — compile-verified
//
#include <hip/hip_runtime.h>

#define DIN 512
#define DOUT 512

typedef __attribute__((ext_vector_type(2))) float v2f;
typedef __attribute__((ext_vector_type(8))) float v8f;

// ---------------- zero workspace (neigh + degs) ----------------
__global__ void sage_zero(float* __restrict__ p, int n) {
    int i = blockIdx.x * blockDim.x + threadIdx.x;
    int stride = gridDim.x * blockDim.x;
    for (; i < n; i += stride) p[i] = 0.0f;
}

// ---------------- in-degree accumulation ----------------
__global__ void sage_degree(const int* __restrict__ dst, float* __restrict__ degs, int n_edges) {
    int e = blockIdx.x * blockDim.x + threadIdx.x;
    if (e < n_edges) atomicAdd(&degs[dst[e]], 1.0f);
}

// ---------------- scatter-add of source features onto dst ----------------
// one block (128 threads) per edge; each thread moves 4 contiguous floats
__global__ void __launch_bounds__(128) sage_scatter(const float* __restrict__ feat,
                                                    const int* __restrict__ src,
                                                    const int* __restrict__ dst,
                                                    float* __restrict__ neigh) {
    int e = blockIdx.x;
    int t = threadIdx.x;                       // 0..127 -> columns t*4 .. t*4+3
    int s = src[e];
    int d = dst[e];
    const float4* fp = (const float4*)(feat + (size_t)s * DIN);
    float4 v = fp[t];
    float* np = neigh + (size_t)d * DIN + t * 4;
    atomicAdd(np + 0, v.x);
    atomicAdd(np + 1, v.y);
    atomicAdd(np + 2, v.z);
    atomicAdd(np + 3, v.w);
}

// ---------------- degs -> 1/max(degs,1) in place ----------------
__global__ void sage_invdeg(float* __restrict__ degs, int n) {
    int i = blockIdx.x * blockDim.x + threadIdx.x;
    if (i < n) degs[i] = 1.0f / fmaxf(degs[i], 1.0f);
}

// ---------------- fused [feat | neigh/deg] @ W + b, ReLU via f32 WMMA ----------------
// wave = one 16(M) x 64(N) strip: 1 A fragment reused by 4 B fragments per k-step.
// 8 waves / block cover the 8 N-groups of one M-tile (DOUT=512 = 8*64).
__global__ void __launch_bounds__(256) sage_gemm(const float* __restrict__ feat,
                                                 const float* __restrict__ neigh,
                                                 const float* __restrict__ invdeg,
                                                 const float* __restrict__ W,
                                                 const float* __restrict__ bias,
                                                 float* __restrict__ out,
                                                 int n_nodes) {
    const int lane   = threadIdx.x & 31;
    const int lwave  = threadIdx.x >> 5;       // 0..7  -> N group
    const int m_tile = blockIdx.x;             // 0..m_tiles-1
    const int nbase  = lwave * 64;
    const int lid    = lane & 15;
    const int half   = lane >> 4;              // lane half selects K offset (+2) per ISA layout
    const int row    = m_tile * 16 + lid;
    const int row_c  = row < n_nodes ? row : (n_nodes - 1);   // clamp reads, no divergence

    v8f acc[4] = {};

    const float invd = invdeg[row_c];
    const float* arow = feat  + (size_t)row_c * DIN;
    const float* nrow = neigh + (size_t)row_c * DIN;

    // ---- K = 0..511 : self features ----
    for (int k = 0; k < DIN; k += 4) {
        const int kk = k + 2 * half;           // lanes 0-15: K=k,k+1 ; lanes 16-31: K=k+2,k+3
        v2f a = *(const v2f*)(arow + kk);
#pragma unroll
        for (int t = 0; t < 4; ++t) {
            const int col = nbase + t * 16 + lid;
            v2f b;
            b.x = W[(size_t)kk       * DOUT + col];
            b.y = W[(size_t)(kk + 1) * DOUT + col];
            acc[t] = __builtin_amdgcn_wmma_f32_16x16x4_f32(
                false, a, false, b, (short)0, acc[t], false, false);
        }
    }

    // ---- K = 512..1023 : normalized neighbor sum ----
    for (int k = 0; k < DIN; k += 4) {
        const int kk = k + 2 * half;
        v2f a = *(const v2f*)(nrow + kk);
        a.x *= invd;
        a.y *= invd;
#pragma unroll
        for (int t = 0; t < 4; ++t) {
            const int col = nbase + t * 16 + lid;
            v2f b;
            b.x = W[(size_t)(DIN + kk)     * DOUT + col];
            b.y = W[(size_t)(DIN + kk + 1) * DOUT + col];
            acc[t] = __builtin_amdgcn_wmma_f32_16x16x4_f32(
                false, a, false, b, (short)0, acc[t], false, false);
        }
    }

    // ---- epilogue: +bias, ReLU, store (C layout: VGPR i -> row i + 8*half) ----
#pragma unroll
    for (int t = 0; t < 4; ++t) {
        const int col = nbase + t * 16 + lid;
        const float bv = bias[col];
#pragma unroll
        for (int i = 0; i < 8; ++i) {
            const int orow = m_tile * 16 + i + 8 * half;
            if (orow < n_nodes) {
                float v = acc[t][i] + bv;
                out[(size_t)orow * DOUT + col] = fmaxf(v, 0.0f);
            }
        }
    }
}

extern "C" void kernel_launch(void* const* d_in, const int* in_sizes, int n_in,
                              void* d_out, int out_size, void* d_ws, size_t ws_size,
                              hipStream_t stream) {
    const float* feat = (const float*)d_in[0];
    const int*   src  = (const int*)d_in[1];
    const int*   dst  = (const int*)d_in[2];
    const float* W    = (const float*)d_in[3];
    const float* bias = (const float*)d_in[4];
    float*       out  = (float*)d_out;

    const int n_nodes = in_sizes[0] / DIN;
    const int n_edges = in_sizes[1];

    float* neigh = (float*)d_ws;                          // n_nodes * 512 f32
    float* degs  = neigh + (size_t)n_nodes * DIN;         // n_nodes f32 (becomes invdeg)

    // 1) zero neigh + degs (workspace is not re-zeroed by the harness)
    const int zn = n_nodes * DIN + n_nodes;
    sage_zero<<<(zn + 255) / 256, 256, 0, stream>>>(neigh, zn);

    // 2) in-degrees
    sage_degree<<<(n_edges + 255) / 256, 256, 0, stream>>>(dst, degs, n_edges);

    // 3) scatter-add source features onto destinations (one block per edge)
    sage_scatter<<<n_edges, 128, 0, stream>>>(feat, src, dst, neigh);

    // 4) degs -> 1/max(degs,1)
    sage_invdeg<<<(n_nodes + 255) / 256, 256, 0, stream>>>(degs, n_nodes);

    // 5) fused concat-GEMM + bias + ReLU with f32 WMMA
    const int m_tiles = (n_nodes + 15) / 16;              // 625 for n=10000
    sage_gemm<<<m_tiles, 256, 0, stream>>>(feat, neigh, degs, W, bias, out, n_nodes);
}